// U3PLloss_88373247082926
// MI455X (gfx1250) — compile-verified
//
#include <hip/hip_runtime.h>
#include <hip/hip_bf16.h>
#include <stdint.h>

#define NCLS 21
#define WDIM 512
#define HWSZ (512 * 512)
#define WIN 25
#define WPIX (WIN * WIN)        // 625
#define WPB 400                 // 20x20 windows per image
#define BORDER_PER_B 12144      // 512*512 - 500*500
#define SSIM_N 12500.0f         // 20 ch * 625 px
#define SSIM_NM1 12499.0f
#define C1F 0.0001f
#define C2F 0.0009f

#if defined(__HIP_DEVICE_COMPILE__) && defined(__gfx1250__) && __has_builtin(__builtin_amdgcn_tensor_load_to_lds)
#define USE_TDM 1
#endif

typedef unsigned int u32x4 __attribute__((ext_vector_type(4)));
typedef int i32x4 __attribute__((ext_vector_type(4)));
typedef int i32x8 __attribute__((ext_vector_type(8)));

__device__ __forceinline__ float block_reduce_256(float v, float* red) {
  const int tid = threadIdx.x;
  red[tid] = v;
  __syncthreads();
#pragma unroll
  for (int s = 128; s >= 1; s >>= 1) {
    if (tid < s) red[tid] += red[tid + s];
    __syncthreads();
  }
  float r = red[0];
  __syncthreads();
  return r;
}

// One block per (b, wy, wx) window. Computes window SSIM stats + partial
// focal/IoU sums over the 25x25 pixel tile.
__global__ __launch_bounds__(256) void win_kernel(const float* __restrict__ pred,
                                                  const int* __restrict__ target,
                                                  float* __restrict__ ws,
                                                  int npart) {
  __shared__ float tile[NCLS * WPIX];   // 52.5 KB: tile[c][py][px] contiguous
  __shared__ float red[256];
  const int tid = threadIdx.x;
  const int blk = blockIdx.x;
  const int b  = blk / WPB;
  const int w  = blk - b * WPB;
  const int wy = w / 20, wx = w - (w / 20) * 20;
  const int y0 = wy * WIN, x0 = wx * WIN;

#ifdef USE_TDM
  if (tid == 0) {
    // Tensor DMA: 3-D tile (x=25, y=25, c=21), fp32, row stride 512,
    // plane stride 262144, into contiguous LDS.
    size_t gaddr = (size_t)(const void*)(pred + (size_t)b * NCLS * HWSZ +
                                         (size_t)y0 * WDIM + x0);
    uint32_t ldsa = (uint32_t)(size_t)(&tile[0]);   // low 32 bits = LDS offset
    u32x4 g0 = { 1u,                                  // count=1, user descriptor
                 ldsa,                                // lds_addr
                 (uint32_t)(gaddr & 0xFFFFFFFFu),     // global_addr[31:0]
                 (uint32_t)((gaddr >> 32) & 0x01FFFFFFu) | (2u << 30) }; // [56:32] | type=2
    i32x8 g1 = { (int)(2u << 16),          // data_size = 4B
                 (int)(512u << 16),        // tensor_dim0[15:0] (=512) @ bits 63:48
                 (int)(512u << 16),        // dim0 hi=0 | tensor_dim1[15:0] (=512)
                 (int)(25u << 16),         // dim1 hi=0 | tile_dim0 = 25
                 (int)(25u | (21u << 16)), // tile_dim1 = 25 | tile_dim2 = 21
                 (int)512,                 // tensor_dim0_stride[31:0] = 512
                 0,                        // stride0 hi | stride1[15:0] (262144 -> 0)
                 4 };                      // tensor_dim1_stride[47:16] = 262144>>16
    i32x4 g2 = { NCLS, 0, 0, 0 };          // tensor_dim2 = 21
    i32x4 g3 = { 0, 0, 0, 0 };
    i32x8 g4 = { 0, 0, 0, 0, 0, 0, 0, 0 }; // extra group (clang-23 form): zero-fill
    __builtin_amdgcn_tensor_load_to_lds(g0, g1, g2, g3, g4, 0);
#if __has_builtin(__builtin_amdgcn_s_wait_tensorcnt)
    __builtin_amdgcn_s_wait_tensorcnt(0);
#else
    asm volatile("s_wait_tensorcnt 0x0" ::: "memory");
#endif
  }
#else
  for (int i = tid; i < NCLS * WPIX; i += 256) {
    int c = i / WPIX, q = i - c * WPIX;
    int py = q / WIN, px = q - py * WIN;
    tile[i] = pred[(size_t)(b * NCLS + c) * HWSZ +
                   (size_t)(y0 + py) * WDIM + (x0 + px)];
  }
#endif
  __syncthreads();

  float s1 = 0.f, s11 = 0.f, s2 = 0.f, s12 = 0.f;
  float inter = 0.f, sump = 0.f, sumt = 0.f, fsum = 0.f, vcnt = 0.f;

  for (int p = tid; p < WPIX; p += 256) {
    const int py = p / WIN, px = p - py * WIN;
    const int y = y0 + py, x = x0 + px;
    const int t = target[(size_t)b * HWSZ + (size_t)y * WDIM + x];
    const int tc = t < 0 ? 0 : (t > NCLS - 1 ? NCLS - 1 : t);

    float v[NCLS];
    float mx = -1e30f;
#pragma unroll
    for (int c = 0; c < NCLS; ++c) { v[c] = tile[c * WPIX + p]; mx = fmaxf(mx, v[c]); }
    float se = 0.f;
#pragma unroll
    for (int c = 0; c < NCLS; ++c) { v[c] = __expf(v[c] - mx); se += v[c]; }
    const float inv = 1.0f / se;

    float sall = 0.f, l1 = 0.f, l11 = 0.f, pt = 0.f;
#pragma unroll
    for (int c = 0; c < NCLS; ++c) {
      float pc = v[c] * inv;
      sall += pc;
      if (c >= 1) { l1 += pc; l11 += pc * pc; }
      pt = (c == tc) ? pc : pt;     // select chain: no dynamic reg indexing
    }
    sump += sall; s1 += l1; s11 += l11;

    const bool valid = (t != 255);
    const float logpt = __logf(pt);
    const float om = 1.0f - pt;
    fsum += valid ? (-(om * om) * logpt) : 0.f;
    vcnt += valid ? 1.f : 0.f;

    const bool hasoh = (t >= 0) && (t <= NCLS - 1);   // one-hot nonzero
    inter += hasoh ? pt : 0.f;
    sumt  += hasoh ? 1.f : 0.f;
    const bool in1 = (t >= 1) && (t <= NCLS - 1);     // channels 1..20 for SSIM
    s2  += in1 ? 1.f : 0.f;
    s12 += in1 ? pt : 0.f;
  }

  s1 = block_reduce_256(s1, red);
  s11 = block_reduce_256(s11, red);
  s2 = block_reduce_256(s2, red);
  s12 = block_reduce_256(s12, red);
  inter = block_reduce_256(inter, red);
  sump = block_reduce_256(sump, red);
  sumt = block_reduce_256(sumt, red);
  fsum = block_reduce_256(fsum, red);
  vcnt = block_reduce_256(vcnt, red);

  if (tid == 0) {
    const float m1 = s1 / SSIM_N, m2 = s2 / SSIM_N;
    const float var1 = (s11 - s1 * s1 / SSIM_N) / SSIM_NM1;
    const float var2 = (s2 - s2 * s2 / SSIM_N) / SSIM_NM1;   // t one-hot: t^2 == t
    const float cov  = (s12 - s1 * s2 / SSIM_N) / SSIM_N;
    const float cc = (2.f * m1 * m2 + C1F) / (m1 * m1 + m2 * m2 + C1F);
    const float ssv = (2.f * cov + C2F) / (var1 + var2 + C2F);
    ws[blk] = cc * ssv;                     // sim[b*400 + w]
    float* P = ws + (gridDim.x);            // partials region (nwinb sims first)
    P[0 * npart + blk] = fsum;
    P[1 * npart + blk] = vcnt;
    P[2 * npart + blk] = inter;
    P[3 * npart + blk] = sump;
    P[4 * npart + blk] = sumt;
  }
}

// Pixels outside the 500x500 window region: focal + IoU contributions only.
__global__ __launch_bounds__(256) void border_kernel(const float* __restrict__ pred,
                                                     const int* __restrict__ target,
                                                     float* __restrict__ ws,
                                                     int nwinb, int npart, int nbpix) {
  __shared__ float red[256];
  const int tid = threadIdx.x;
  const int i = blockIdx.x * 256 + tid;
  float fsum = 0.f, vcnt = 0.f, inter = 0.f, sump = 0.f, sumt = 0.f;

  if (i < nbpix) {
    const int b = i / BORDER_PER_B;
    const int j = i - b * BORDER_PER_B;
    int y, x;
    if (j < 6144) { y = 500 + (j >> 9); x = j & 511; }            // rows 500..511
    else { int k = j - 6144; y = k / 12; x = 500 + (k - (k / 12) * 12); }
    const float* base = pred + (size_t)b * NCLS * HWSZ + (size_t)y * WDIM + x;
    const int t = target[(size_t)b * HWSZ + (size_t)y * WDIM + x];
    const int tc = t < 0 ? 0 : (t > NCLS - 1 ? NCLS - 1 : t);

    float v[NCLS];
    float mx = -1e30f;
#pragma unroll
    for (int c = 0; c < NCLS; ++c) { v[c] = base[(size_t)c * HWSZ]; mx = fmaxf(mx, v[c]); }
    float se = 0.f;
#pragma unroll
    for (int c = 0; c < NCLS; ++c) { v[c] = __expf(v[c] - mx); se += v[c]; }
    const float inv = 1.0f / se;
    float sall = 0.f, pt = 0.f;
#pragma unroll
    for (int c = 0; c < NCLS; ++c) {
      float pc = v[c] * inv;
      sall += pc;
      pt = (c == tc) ? pc : pt;
    }
    sump = sall;
    const bool valid = (t != 255);
    const float logpt = __logf(pt);
    const float om = 1.0f - pt;
    fsum = valid ? (-(om * om) * logpt) : 0.f;
    vcnt = valid ? 1.f : 0.f;
    const bool hasoh = (t >= 0) && (t <= NCLS - 1);
    inter = hasoh ? pt : 0.f;
    sumt  = hasoh ? 1.f : 0.f;
  }

  fsum = block_reduce_256(fsum, red);
  vcnt = block_reduce_256(vcnt, red);
  inter = block_reduce_256(inter, red);
  sump = block_reduce_256(sump, red);
  sumt = block_reduce_256(sumt, red);

  if (tid == 0) {
    float* P = ws + nwinb;
    const int slot = nwinb + blockIdx.x;
    P[0 * npart + slot] = fsum;
    P[1 * npart + slot] = vcnt;
    P[2 * npart + slot] = inter;
    P[3 * npart + slot] = sump;
    P[4 * npart + slot] = sumt;
  }
}

__global__ __launch_bounds__(256) void final_kernel(const float* __restrict__ ws,
                                                    float* __restrict__ out,
                                                    int out_size, int B, int npart) {
  __shared__ float red[256];
  __shared__ float gsum[5];
  __shared__ float bt[8];
  const int tid = threadIdx.x;
  const int nwinb = B * WPB;
  const float* P = ws + nwinb;

  for (int q = 0; q < 5; ++q) {
    float a = 0.f;
    for (int k = tid; k < npart; k += 256) a += P[q * npart + k];
    a = block_reduce_256(a, red);
    if (tid == 0) gsum[q] = a;
  }

  float tb[8];
#pragma unroll
  for (int b = 0; b < 8; ++b) tb[b] = 0.f;
  for (int w = tid; w < WPB; w += 256) {
    bool ok = true;
    float sv[8];
#pragma unroll
    for (int b = 0; b < 8; ++b) {
      float s = (b < B) ? ws[b * WPB + w] : 0.f;
      sv[b] = s;
      ok = ok && (s == s);               // ~any(isnan) across batch
    }
#pragma unroll
    for (int b = 0; b < 8; ++b) tb[b] += ok ? sv[b] : 0.f;
  }
#pragma unroll
  for (int b = 0; b < 8; ++b) {
    float a = block_reduce_256(tb[b], red);
    if (tid == 0) bt[b] = a;
  }
  __syncthreads();

  if (tid == 0) {
    const float floss = gsum[0] / fmaxf(gsum[1], 1.f);
    const float uni = gsum[3] + gsum[4] - gsum[2];
    const float iou = 1.f - (gsum[2] + 0.1f) / (uni + 0.1f);
    for (int b = 0; b < 8; ++b) {
      if (b < B && b < out_size) {
        const float simb = bt[b] / ((float)WPB + 0.0001f);
        out[b] = floss + iou + (1.f - simb);
      }
    }
  }
}

extern "C" void kernel_launch(void* const* d_in, const int* in_sizes, int n_in,
                              void* d_out, int out_size, void* d_ws, size_t ws_size,
                              hipStream_t stream) {
  const float* pred = (const float*)d_in[0];
  const int* target = (const int*)d_in[1];
  float* out = (float*)d_out;
  float* ws = (float*)d_ws;

  const int B = in_sizes[0] / (NCLS * HWSZ);      // 8
  const int nwinb = B * WPB;                      // 3200
  const int nbpix = B * BORDER_PER_B;             // 97152
  const int nbblk = (nbpix + 255) / 256;          // 380
  const int npart = nwinb + nbblk;                // 3580

  win_kernel<<<nwinb, 256, 0, stream>>>(pred, target, ws, npart);
  border_kernel<<<nbblk, 256, 0, stream>>>(pred, target, ws, nwinb, npart, nbpix);
  final_kernel<<<1, 256, 0, stream>>>(ws, out, out_size, B, npart);
}